// BlockPiecewiseLinear_3831110828574
// MI455X (gfx1250) — compile-verified
//
#include <hip/hip_runtime.h>

// Problem constants (match reference)
#define BPL_B 4096
#define BPL_F 512
#define BPL_K 32          // knots; K+1 = 33 slopes

constexpr int TPB = 128;          // threads per block = pairs per block (4 waves of 32)
constexpr int XS_STRIDE = 33;     // padded LDS row stride (DWORDs) for x : gcd(33,64)=1 -> conflict-free
constexpr int SL_STRIDE = 33;     // slope rows are naturally 33 floats: contiguous AND conflict-free

// Address-space-qualified pointer types (probe-verified casts)
typedef __attribute__((address_space(1))) int* bpl_gptr;  // global
typedef __attribute__((address_space(3))) int* bpl_lptr;  // LDS

// ---- Path selection: TDM (tensor DMA) > per-lane async-to-LDS > plain ds ----
#if defined(__HIP_DEVICE_COMPILE__) && defined(__gfx1250__) && \
    __has_builtin(__builtin_amdgcn_tensor_load_to_lds) && \
    __has_builtin(__builtin_amdgcn_s_wait_tensorcnt)
#define BPL_TDM 1
#else
#define BPL_TDM 0
#endif

#if !BPL_TDM && defined(__HIP_DEVICE_COMPILE__) && defined(__gfx1250__) && \
    __has_builtin(__builtin_amdgcn_global_load_async_to_lds_b32)
#define BPL_ASYNC 1
#else
#define BPL_ASYNC 0
#endif

#if BPL_TDM
typedef unsigned int bpl_v4u __attribute__((ext_vector_type(4)));
typedef int bpl_v8i __attribute__((ext_vector_type(8)));
typedef int bpl_v4i __attribute__((ext_vector_type(4)));
#endif

#if BPL_ASYNC
__device__ __forceinline__ void bpl_async_b32(const float* g, float* l) {
  __builtin_amdgcn_global_load_async_to_lds_b32((bpl_gptr)g, (bpl_lptr)l, 0, 0);
}
__device__ __forceinline__ void bpl_wait_async() {
#if __has_builtin(__builtin_amdgcn_s_wait_asynccnt)
  __builtin_amdgcn_s_wait_asynccnt(0);
#else
  asm volatile("s_wait_asynccnt 0" ::: "memory");
#endif
}
#endif

__device__ __forceinline__ float softplus_eps(float x) {
  // softplus(x) + 1e-3, overflow-safe; abs error far below the 1e-3 floor
  return fmaxf(x, 0.0f) + __logf(1.0f + __expf(-fabsf(x))) + 1.0e-3f;
}

__global__ __launch_bounds__(TPB) void bpl_kernel(
    const float* __restrict__ g_in,   // [B*F]
    const float* __restrict__ g_x,    // [B*F, K]
    const float* __restrict__ g_s,    // [B*F, K+1]
    const float* __restrict__ g_yb,   // [F]
    float* __restrict__ g_out)        // [2*B*F] : outputs then slope_sel
{
  __shared__ float lds_x[TPB * XS_STRIDE];
  __shared__ float lds_s[TPB * SL_STRIDE];

  const int t = threadIdx.x;
  const size_t pair = (size_t)blockIdx.x * TPB + t;
  const float* xsrc = g_x + (size_t)blockIdx.x * TPB * BPL_K;
  const float* ssrc = g_s + (size_t)blockIdx.x * TPB * (BPL_K + 1);

  // ---------------- Stage tiles into LDS ----------------------------------------
#if BPL_TDM
  if (t < 32) {  // wave-uniform: only wave 0 issues the DMAs (TDM ignores EXEC)
    unsigned lx = (unsigned)(unsigned long long)(bpl_lptr)&lds_x[0];
    unsigned ls = (unsigned)(unsigned long long)(bpl_lptr)&lds_s[0];
    unsigned long long gx = (unsigned long long)xsrc;
    unsigned long long gs = (unsigned long long)ssrc;

    // Shared group2/group3: 2D tiles; unused higher dims set to 1/0 (benign)
    bpl_v4i g2 = { 1, 1, 0, 0 };               // tensor_dim2=1, tensor_dim3=1
    bpl_v4i g3 = { 0, (int)(1u << 16), 0, 0 }; // tensor_dim4=1
    bpl_v8i g4 = { 0, 0, 0, 0, 0, 0, 0, 0 };   // 5th arg on clang-23 builtin: zero-filled

    // --- x tile: 128 rows x 32 f32, TDM inserts 1-DWORD pad per 32 DWORDs ------
    // group1 d0: data_size=2(4B)<<16 | pad_enable<<20 | pad_interval(code4=32dw)<<22
    bpl_v4u g0x = { 1u, lx, (unsigned)gx, (unsigned)(gx >> 32) | (2u << 30) };
    bpl_v8i g1x = { (int)0x01120000u,
                    (int)(32u  << 16),   // tensor_dim0 = 32
                    (int)(128u << 16),   // tensor_dim1 = 128
                    (int)(32u  << 16),   // tile_dim0   = 32
                    (int)128u,           // tile_dim1   = 128, tile_dim2 = 0
                    (int)32u,            // tensor_dim0_stride = 32
                    0, 0 };
    __builtin_amdgcn_tensor_load_to_lds(g0x, g1x, g2, g3, g4, 0);

    // --- slope tile: flat contiguous 128*33 = 4224 f32, no padding -------------
    bpl_v4u g0s = { 1u, ls, (unsigned)gs, (unsigned)(gs >> 32) | (2u << 30) };
    bpl_v8i g1s = { (int)0x00020000u,    // data_size=2 (4B), no pad
                    (int)(4224u << 16),  // tensor_dim0 = 4224
                    (int)(1u    << 16),  // tensor_dim1 = 1
                    (int)(4224u << 16),  // tile_dim0   = 4224
                    (int)1u,             // tile_dim1   = 1
                    (int)4224u,          // tensor_dim0_stride = 4224
                    0, 0 };
    __builtin_amdgcn_tensor_load_to_lds(g0s, g1s, g2, g3, g4, 0);

    __builtin_amdgcn_s_wait_tensorcnt(0);
  }
#else
  {
    // x tile: TPB*32 floats, pair-major; scatter into padded rows
#pragma unroll
    for (int i = 0; i < BPL_K; ++i) {
      int f = i * TPB + t;
      int p = f >> 5;
      int e = f & 31;
#if BPL_ASYNC
      bpl_async_b32(xsrc + f, &lds_x[p * XS_STRIDE + e]);
#else
      lds_x[p * XS_STRIDE + e] = __builtin_nontemporal_load(xsrc + f);
#endif
    }
    // slope tile: contiguous copy (stride 33 is already conflict-free)
#pragma unroll
    for (int i = 0; i < BPL_K + 1; ++i) {
      int f = i * TPB + t;
#if BPL_ASYNC
      bpl_async_b32(ssrc + f, &lds_s[f]);
#else
      lds_s[f] = __builtin_nontemporal_load(ssrc + f);
#endif
    }
  }
#if BPL_ASYNC
  bpl_wait_async();
#endif
#endif
  __syncthreads();

  // ---------------- Per-thread: load row, bitonic sort in registers -------------
  float xs[BPL_K];
  {
    const float* xrow = &lds_x[t * XS_STRIDE];
#pragma unroll
    for (int i = 0; i < BPL_K; ++i) xs[i] = xrow[i];
  }

  // Fully unrolled 32-element bitonic sort (ascending); constant indices -> VGPRs
#pragma unroll
  for (int k = 2; k <= BPL_K; k <<= 1) {
#pragma unroll
    for (int j = k >> 1; j > 0; j >>= 1) {
#pragma unroll
      for (int i = 0; i < BPL_K; ++i) {
        int ixj = i ^ j;
        if (ixj > i) {
          bool up = (i & k) == 0;
          float a = xs[i], b = xs[ixj];
          float lo = fminf(a, b), hi = fmaxf(a, b);
          xs[i]   = up ? lo : hi;
          xs[ixj] = up ? hi : lo;
        }
      }
    }
  }

  // ---------------- Fused softplus + cumsum + searchsorted + select -------------
  const float xin = g_in[pair];
  const float yb  = g_yb[pair & (BPL_F - 1)];
  const float* srow = &lds_s[t * SL_STRIDE];

  float s_sel = softplus_eps(srow[0]);   // slope_c[0] (slope_idx==0 case)
  float y     = xs[0] + yb;              // y_pos[0]
  float x_sel = xs[0];
  float y_sel = y;

#pragma unroll
  for (int i = 0; i < BPL_K; ++i) {
    float sp = softplus_eps(srow[i + 1]);        // slope_c[i+1]
    bool ge = (xin >= xs[i]);                    // contributes to slope_idx
    if (ge) { x_sel = xs[i]; y_sel = y; s_sel = sp; }
    if (i < BPL_K - 1) y += (xs[i + 1] - xs[i]) * sp;  // cumsum of delta_y
  }

  float out = y_sel + (xin - x_sel) * s_sel;
  __builtin_nontemporal_store(out,  g_out + pair);
  __builtin_nontemporal_store(s_sel, g_out + (size_t)BPL_B * BPL_F + pair);
}

extern "C" void kernel_launch(void* const* d_in, const int* in_sizes, int n_in,
                              void* d_out, int out_size, void* d_ws, size_t ws_size,
                              hipStream_t stream) {
  const float* g_in = (const float*)d_in[0];   // inputs [B,F]
  const float* g_x  = (const float*)d_in[1];   // x_pos  [B,F,K]
  const float* g_s  = (const float*)d_in[2];   // slope  [B,F,K+1]
  const float* g_yb = (const float*)d_in[3];   // y_bias [F,1]
  float* out = (float*)d_out;                  // [2*B*F]

  const int pairs = BPL_B * BPL_F;
  dim3 grid(pairs / TPB), block(TPB);
  hipLaunchKernelGGL(bpl_kernel, grid, block, 0, stream, g_in, g_x, g_s, g_yb, out);
}